// ResnetA_63969242906670
// MI455X (gfx1250) — compile-verified
//
#include <hip/hip_runtime.h>
#include <hip/hip_bf16.h>
#include <math.h>

#define NPTS   8192
#define IN_C   256
#define OUT_C  256
#define SCH    64          // shortcut channels (IN_C/4)
#define KNN    32
#define KP     15
#define RADIUS 0.12f
#define RADIUS2 (RADIUS * RADIUS)
#define EXTENT 0.096f      // RADIUS * 2.0 / 2.5
#define INV_EXTENT (1.0f / EXTENT)
#define NEG_SLOPE 0.1f
#define BN_EPS 1e-5f
#define AGST (KP * SCH + 4)   // 964: padded LDS row stride (960 % 64 == 0 would 16-way conflict)

typedef __attribute__((ext_vector_type(2))) float v2f;
typedef __attribute__((ext_vector_type(8))) float v8f;

// D = A(16x4) * B(4x16) + C, fp32, wave32.
// Layouts (ISA 7.12.2): A[M][K]: lane=(K>>1)*16+M, vgpr=K&1.
//                       B[K][N]: lane=(K>>1)*16+N, vgpr=K&1.
//                       D[M][N]: lane=(M>=8?16:0)+N, vgpr=M%8.
__device__ __forceinline__ v8f wmma_f32_16x16x4(v2f a, v2f b, v8f c) {
  return __builtin_amdgcn_wmma_f32_16x16x4_f32(false, a, false, b, (short)0, c,
                                               false, false);
}

// ---------------------------------------------------------------------------
// h[n,j] = sum_c x[n,c] * W[c,j] + bias[j];  x:[NPTS,IN_C], W:[IN_C,SCH]
// grid NPTS/16 blocks x 128 threads (4 waves); wave w owns N-tile j0 = 16w.
// ---------------------------------------------------------------------------
__global__ void __launch_bounds__(128)
gemm_in_kernel(const float* __restrict__ x, const float* __restrict__ W,
               const float* __restrict__ bias, float* __restrict__ h) {
  const int lane = threadIdx.x & 31;
  const int wid  = threadIdx.x >> 5;
  const int m0   = blockIdx.x * 16;
  const int j0   = wid * 16;
  const int row  = lane & 15;
  const int hl   = lane >> 4;
  v8f acc = {};
  const float* ap = x + (size_t)(m0 + row) * IN_C + 2 * hl;
  #pragma unroll 4
  for (int k0 = 0; k0 < IN_C; k0 += 4) {
    v2f a, b;
    a.x = ap[k0 + 0];
    a.y = ap[k0 + 1];
    b.x = W[(size_t)(k0 + 2 * hl    ) * SCH + j0 + row];
    b.y = W[(size_t)(k0 + 2 * hl + 1) * SCH + j0 + row];
    acc = wmma_f32_16x16x4(a, b, acc);
  }
  const float bb = bias[j0 + row];
  #pragma unroll
  for (int v = 0; v < 8; ++v)
    h[(size_t)(m0 + v + hl * 8) * SCH + j0 + row] = acc[v] + bb;
}

// ---------------------------------------------------------------------------
// Brute-force kNN: thread-per-point, LDS-tiled candidates, register top-32
// via max-replace with fully unrolled select chains (stays in VGPRs).
// ---------------------------------------------------------------------------
__global__ void __launch_bounds__(256)
knn_kernel(const float* __restrict__ coords, int* __restrict__ nb) {
  __shared__ float sx[256], sy[256], sz[256];
  const int n = blockIdx.x * 256 + threadIdx.x;
  const float px = coords[n * 3 + 0];
  const float py = coords[n * 3 + 1];
  const float pz = coords[n * 3 + 2];
  float bd[KNN]; int bi[KNN];
  #pragma unroll
  for (int i = 0; i < KNN; ++i) { bd[i] = 3.0e38f; bi[i] = 0; }
  float worst = 3.0e38f; int wslot = 0;
  for (int t = 0; t < NPTS; t += 256) {
    __syncthreads();
    const int j = t + threadIdx.x;
    sx[threadIdx.x] = coords[j * 3 + 0];
    sy[threadIdx.x] = coords[j * 3 + 1];
    sz[threadIdx.x] = coords[j * 3 + 2];
    __syncthreads();
    for (int jj = 0; jj < 256; ++jj) {
      const float dx = sx[jj] - px, dy = sy[jj] - py, dz = sz[jj] - pz;
      const float d2 = dx * dx + dy * dy + dz * dz;
      if (d2 < worst) {
        #pragma unroll
        for (int i = 0; i < KNN; ++i)
          if (i == wslot) { bd[i] = d2; bi[i] = t + jj; }
        worst = -1.0f;
        #pragma unroll
        for (int i = 0; i < KNN; ++i)
          if (bd[i] > worst) { worst = bd[i]; wslot = i; }
      }
    }
  }
  #pragma unroll
  for (int i = 0; i < KNN; ++i) nb[n * KNN + i] = bi[i];
}

// ---------------------------------------------------------------------------
// KPConv, fused per 16-point block.
// Phase A: 8 waves (2 points each) build agg[16][KP*SCH] in LDS.
//   Influence for kernel point p computed in lane p, broadcast via __shfl.
// Phase B: 4 waves run h2[16,64] = agg[16,960] @ Wkp_flat[960,64] with WMMA.
// ---------------------------------------------------------------------------
__global__ void __launch_bounds__(256)
kpconv_kernel(const float* __restrict__ h, const int* __restrict__ nbidx,
              const float* __restrict__ coords, const float* __restrict__ kpts,
              const float* __restrict__ Wkp, float* __restrict__ h2) {
  __shared__ float agg[16 * AGST];
  const int lane = threadIdx.x & 31;
  const int wid  = threadIdx.x >> 5;
  const int n0   = blockIdx.x * 16;

  float kx = 0.f, ky = 0.f, kz = 0.f;
  if (lane < KP) {
    kx = kpts[lane * 3 + 0];
    ky = kpts[lane * 3 + 1];
    kz = kpts[lane * 3 + 2];
  }

  for (int s = 0; s < 2; ++s) {
    const int m = wid * 2 + s;
    const int n = n0 + m;
    const float cx = coords[n * 3 + 0];
    const float cy = coords[n * 3 + 1];
    const float cz = coords[n * 3 + 2];
    float acc0[KP], acc1[KP];
    #pragma unroll
    for (int p = 0; p < KP; ++p) { acc0[p] = 0.f; acc1[p] = 0.f; }
    for (int k = 0; k < KNN; ++k) {
      const int j = nbidx[n * KNN + k];
      const float rx = coords[j * 3 + 0] - cx;
      const float ry = coords[j * 3 + 1] - cy;
      const float rz = coords[j * 3 + 2] - cz;
      const float r2 = rx * rx + ry * ry + rz * rz;
      const float mask = (r2 <= RADIUS2) ? 1.0f : 0.0f;
      const float f0 = h[(size_t)j * SCH + lane];
      const float f1 = h[(size_t)j * SCH + lane + 32];
      const float dx = rx - kx, dy = ry - ky, dz = rz - kz;
      const float dist = sqrtf(dx * dx + dy * dy + dz * dz);
      const float infl = fmaxf(1.0f - dist * INV_EXTENT, 0.0f) * mask;
      #pragma unroll
      for (int p = 0; p < KP; ++p) {
        const float ip = __shfl(infl, p, 32);
        acc0[p] = fmaf(ip, f0, acc0[p]);
        acc1[p] = fmaf(ip, f1, acc1[p]);
      }
    }
    #pragma unroll
    for (int p = 0; p < KP; ++p) {
      agg[m * AGST + p * SCH + lane]      = acc0[p];
      agg[m * AGST + p * SCH + lane + 32] = acc1[p];
    }
  }
  __syncthreads();

  if (wid < 4) {                 // whole waves -> EXEC all ones inside
    const int row = lane & 15;
    const int hl  = lane >> 4;
    const int j0  = wid * 16;
    v8f acc = {};
    #pragma unroll 4
    for (int k0 = 0; k0 < KP * SCH; k0 += 4) {
      v2f a, b;
      a.x = agg[row * AGST + k0 + 2 * hl + 0];
      a.y = agg[row * AGST + k0 + 2 * hl + 1];
      b.x = Wkp[(size_t)(k0 + 2 * hl    ) * SCH + j0 + row];
      b.y = Wkp[(size_t)(k0 + 2 * hl + 1) * SCH + j0 + row];
      acc = wmma_f32_16x16x4(a, b, acc);
    }
    #pragma unroll
    for (int v = 0; v < 8; ++v)
      h2[(size_t)(n0 + v + hl * 8) * SCH + j0 + row] = acc[v];
  }
}

// ---------------------------------------------------------------------------
// y[n,d] = sum_c h2[n,c] * W[c,d] + bias[d];  [NPTS,SCH] x [SCH,OUT_C]
// 8 waves per block, each wave does 2 N-tiles.
// ---------------------------------------------------------------------------
__global__ void __launch_bounds__(256)
gemm_out_kernel(const float* __restrict__ hh, const float* __restrict__ W,
                const float* __restrict__ bias, float* __restrict__ y) {
  const int lane = threadIdx.x & 31;
  const int wid  = threadIdx.x >> 5;
  const int m0   = blockIdx.x * 16;
  const int row  = lane & 15;
  const int hl   = lane >> 4;
  const float* ap = hh + (size_t)(m0 + row) * SCH + 2 * hl;
  #pragma unroll
  for (int jt = 0; jt < 2; ++jt) {
    const int j0 = (wid + jt * 8) * 16;
    v8f acc = {};
    #pragma unroll 4
    for (int k0 = 0; k0 < SCH; k0 += 4) {
      v2f a, b;
      a.x = ap[k0 + 0];
      a.y = ap[k0 + 1];
      b.x = W[(size_t)(k0 + 2 * hl    ) * OUT_C + j0 + row];
      b.y = W[(size_t)(k0 + 2 * hl + 1) * OUT_C + j0 + row];
      acc = wmma_f32_16x16x4(a, b, acc);
    }
    const float bb = bias[j0 + row];
    #pragma unroll
    for (int v = 0; v < 8; ++v)
      y[(size_t)(m0 + v + hl * 8) * OUT_C + j0 + row] = acc[v] + bb;
  }
}

// ---------------------------------------------------------------------------
// Deterministic BN: per-block partial sums -> single-block finalize -> apply.
// ---------------------------------------------------------------------------
__global__ void __launch_bounds__(256)
bn_partial_kernel(const float* __restrict__ y, float* __restrict__ psum,
                  float* __restrict__ psq) {
  const int ch = threadIdx.x;                 // 0..255
  const int nbase = blockIdx.x * (NPTS / 32); // 32 blocks
  float s = 0.f, q = 0.f;
  for (int i = 0; i < NPTS / 32; ++i) {
    const float v = y[(size_t)(nbase + i) * OUT_C + ch];
    s += v; q += v * v;
  }
  psum[blockIdx.x * OUT_C + ch] = s;
  psq [blockIdx.x * OUT_C + ch] = q;
}

__global__ void __launch_bounds__(256)
bn_final_kernel(const float* __restrict__ psum, const float* __restrict__ psq,
                float* __restrict__ stat) {
  const int ch = threadIdx.x;
  float s = 0.f, q = 0.f;
  for (int b = 0; b < 32; ++b) { s += psum[b * OUT_C + ch]; q += psq[b * OUT_C + ch]; }
  const float mu = s * (1.0f / NPTS);
  stat[ch]         = mu;
  stat[OUT_C + ch] = q * (1.0f / NPTS) - mu * mu;   // E[y^2]-mu^2
}

__global__ void __launch_bounds__(256)
bn_apply_kernel(float* __restrict__ y, const float* __restrict__ stat,
                const float* __restrict__ gamma, const float* __restrict__ beta) {
  const int i  = blockIdx.x * 256 + threadIdx.x;
  const int ch = i & (OUT_C - 1);
  const float mu  = stat[ch];
  const float var = stat[OUT_C + ch];
  const float v = (y[i] - mu) * rsqrtf(var + BN_EPS) * gamma[ch] + beta[ch];
  y[i] = v > 0.f ? v : NEG_SLOPE * v;
}

__global__ void __launch_bounds__(256)
copy_coords_kernel(const float* __restrict__ a, const float* __restrict__ b,
                   float* __restrict__ o) {
  const int i = blockIdx.x * blockDim.x + threadIdx.x;
  if (i < NPTS * 3) { o[i] = a[i]; o[NPTS * 3 + i] = b[i]; }
}

// ---------------------------------------------------------------------------
extern "C" void kernel_launch(void* const* d_in, const int* in_sizes, int n_in,
                              void* d_out, int out_size, void* d_ws, size_t ws_size,
                              hipStream_t stream) {
  const float* src  = (const float*)d_in[0];
  const float* tgt  = (const float*)d_in[1];
  const float* srcc = (const float*)d_in[2];
  const float* tgtc = (const float*)d_in[3];
  const float* W_in = (const float*)d_in[4];
  const float* b_in = (const float*)d_in[5];
  const float* kpts = (const float*)d_in[6];
  const float* W_kp = (const float*)d_in[7];   // [KP,SCH,SCH] == flat [960,64]
  const float* W_out= (const float*)d_in[8];
  const float* b_out= (const float*)d_in[9];
  const float* gamma= (const float*)d_in[10];
  const float* beta = (const float*)d_in[11];
  float* out = (float*)d_out;

  // Workspace layout (floats): h | h2 | psum | psq | stat | idx(int)
  float* ws   = (float*)d_ws;
  float* h    = ws;
  float* h2   = h    + (size_t)NPTS * SCH;
  float* psum = h2   + (size_t)NPTS * SCH;
  float* psq  = psum + (size_t)32 * OUT_C;
  float* stat = psq  + (size_t)32 * OUT_C;
  int*   nb   = (int*)(stat + 2 * OUT_C);      // NPTS*KNN ints

  for (int c = 0; c < 2; ++c) {
    const float* x  = c ? tgt  : src;
    const float* co = c ? tgtc : srcc;
    float* yout = out + (size_t)c * NPTS * OUT_C;

    gemm_in_kernel <<<NPTS / 16, 128, 0, stream>>>(x, W_in, b_in, h);
    knn_kernel     <<<NPTS / 256, 256, 0, stream>>>(co, nb);
    kpconv_kernel  <<<NPTS / 16, 256, 0, stream>>>(h, nb, co, kpts, W_kp, h2);
    gemm_out_kernel<<<NPTS / 16, 256, 0, stream>>>(h2, W_out, b_out, yout);
    bn_partial_kernel<<<32, 256, 0, stream>>>(yout, psum, psq);
    bn_final_kernel  <<<1, 256, 0, stream>>>(psum, psq, stat);
    bn_apply_kernel  <<<(NPTS * OUT_C) / 256, 256, 0, stream>>>(yout, stat, gamma, beta);
  }
  copy_coords_kernel<<<(NPTS * 3 + 255) / 256, 256, 0, stream>>>(
      srcc, tgtc, out + (size_t)2 * NPTS * OUT_C);

  (void)in_sizes; (void)n_in; (void)out_size; (void)ws_size;
}